// GraphSAGE_11819749998735
// MI455X (gfx1250) — compile-verified
//
#include <hip/hip_runtime.h>
#include <cstdint>

#define NN   100000   // nodes
#define DIM  64       // EMB == HID == 64
#define NT   6250     // node tiles of 16 (16*6250 == 100000 exactly)

typedef float v2f __attribute__((ext_vector_type(2)));
typedef float v8f __attribute__((ext_vector_type(8)));

// ---------------- zero ----------------
__global__ void zero_f32(float* __restrict__ p, int n) {
    int i = blockIdx.x * blockDim.x + threadIdx.x;
    if (i < n) p[i] = 0.0f;
}

// ---------------- degree: deg[dst[e]] += 1 ----------------
__global__ void deg_kernel(const long long* __restrict__ dst,
                           float* __restrict__ deg, int E) {
    int e = blockIdx.x * blockDim.x + threadIdx.x;
    if (e < E) {
        int d = (int)dst[e];
        unsafeAtomicAdd(&deg[d], 1.0f);
    }
}

// ---------------- scatter-add: agg[dst[e]][:] += x[src[e]][:] ----------------
// one thread per (edge, 4-float chunk); 16 consecutive threads cover one edge's
// 64 floats -> coalesced 128b gather loads, native f32 atomics on the scatter.
__global__ void scatter_kernel(const float* __restrict__ x,
                               const long long* __restrict__ src,
                               const long long* __restrict__ dst,
                               float* __restrict__ agg, int E) {
    long t = (long)blockIdx.x * blockDim.x + threadIdx.x;
    if (t >= (long)E * 16) return;
    int e = (int)(t >> 4);
    int c = (int)(t & 15) << 2;
    int s = (int)src[e];
    int d = (int)dst[e];
    float4 v = *(const float4*)(x + (long)s * DIM + c);
    float* o = agg + (long)d * DIM + c;
    unsafeAtomicAdd(o + 0, v.x);
    unsafeAtomicAdd(o + 1, v.y);
    unsafeAtomicAdd(o + 2, v.z);
    unsafeAtomicAdd(o + 3, v.w);
}

// ---------------- SAGE dense step via V_WMMA_F32_16X16X4_F32 ----------------
// out[16-node tile] = (agg/deg) @ wl^T + bl + x @ wr^T   (optional ReLU)
// One wave handles one 16-row node tile, looping over 4 column tiles; K=64 in
// 16 steps of 4. A/B fragments are contiguous float2 loads per the ISA layout.
template <bool RELU>
__global__ __launch_bounds__(256)
void sage_gemm(const float* __restrict__ agg, const float* __restrict__ deg,
               const float* __restrict__ x,
               const float* __restrict__ wl, const float* __restrict__ bl,
               const float* __restrict__ wr,
               float* __restrict__ out) {
    const int lane = threadIdx.x & 31;
    const int wave = threadIdx.x >> 5;
    const int tile = blockIdx.x * 8 + wave;
    if (tile >= NT) return;                      // wave-uniform: EXEC stays full

    const int m0   = tile * 16;
    const int row  = m0 + (lane & 15);           // A-fragment row for this lane
    const int kh   = (lane >> 4) * 2;            // k sub-offset: 0 or 2
    const float inv = 1.0f / fmaxf(deg[row], 1.0f);

    const float* aggRow = agg + (long)row * DIM;
    const float* xRow   = x   + (long)row * DIM;

    #pragma unroll
    for (int nt = 0; nt < 4; ++nt) {
        const int n0   = nt * 16;
        const int hcol = n0 + (lane & 15);       // B-fragment column / D column
        const float* wlRow = wl + (long)hcol * DIM;
        const float* wrRow = wr + (long)hcol * DIM;

        v8f c = {};
        #pragma unroll
        for (int k0 = 0; k0 < DIM; k0 += 4) {
            v2f aA = *(const v2f*)(aggRow + k0 + kh);
            aA.x *= inv; aA.y *= inv;
            v2f bA = *(const v2f*)(wlRow + k0 + kh);
            c = __builtin_amdgcn_wmma_f32_16x16x4_f32(
                    false, aA, false, bA, (short)0, c, false, false);

            v2f aX = *(const v2f*)(xRow + k0 + kh);
            v2f bX = *(const v2f*)(wrRow + k0 + kh);
            c = __builtin_amdgcn_wmma_f32_16x16x4_f32(
                    false, aX, false, bX, (short)0, c, false, false);
        }

        const float bias  = bl[hcol];
        const int   mbase = m0 + 8 * (lane >> 4);
        #pragma unroll
        for (int r = 0; r < 8; ++r) {
            float v = c[r] + bias;
            if (RELU) v = fmaxf(v, 0.0f);
            out[(long)(mbase + r) * DIM + hcol] = v;
        }
    }
}

// ---------------- row-wise L2 normalize (in place) ----------------
// one wave per node: 32 lanes x float2, shuffle reduction.
__global__ void l2norm_kernel(float* __restrict__ out, int n) {
    int node = blockIdx.x * (blockDim.x >> 5) + (threadIdx.x >> 5);
    int lane = threadIdx.x & 31;
    if (node >= n) return;
    float* p = out + (long)node * DIM + lane * 2;
    float2 v = *(float2*)p;
    float s = v.x * v.x + v.y * v.y;
    #pragma unroll
    for (int off = 16; off > 0; off >>= 1)
        s += __shfl_xor(s, off, 32);
    float invn = 1.0f / fmaxf(sqrtf(s), 1e-12f);
    v.x *= invn; v.y *= invn;
    *(float2*)p = v;
}

extern "C" void kernel_launch(void* const* d_in, const int* in_sizes, int n_in,
                              void* d_out, int out_size, void* d_ws, size_t ws_size,
                              hipStream_t stream) {
    const float*     emb  = (const float*)d_in[0];
    const float*     wl0  = (const float*)d_in[1];
    const float*     bl0  = (const float*)d_in[2];
    const float*     wr0  = (const float*)d_in[3];
    const float*     wl1  = (const float*)d_in[4];
    const float*     bl1  = (const float*)d_in[5];
    const float*     wr1  = (const float*)d_in[6];
    const long long* eidx = (const long long*)d_in[7];
    const int E = in_sizes[7] / 2;
    const long long* src = eidx;
    const long long* dst = eidx + E;

    float* out = (float*)d_out;

    // workspace layout: [agg: N*64][deg: N][h: N*64]
    float* agg = (float*)d_ws;
    float* deg = agg + (long)NN * DIM;
    float* h   = deg + NN;

    const int THREADS = 256;
    const int nZero   = NN * DIM + NN;                 // agg + deg contiguous
    const long nScat  = (long)E * 16;

    // ---- layer 0 ----
    zero_f32<<<(nZero + THREADS - 1) / THREADS, THREADS, 0, stream>>>(agg, nZero);
    deg_kernel<<<(E + THREADS - 1) / THREADS, THREADS, 0, stream>>>(dst, deg, E);
    scatter_kernel<<<(int)((nScat + THREADS - 1) / THREADS), THREADS, 0, stream>>>(
        emb, src, dst, agg, E);
    sage_gemm<true><<<(NT + 7) / 8, THREADS, 0, stream>>>(
        agg, deg, emb, wl0, bl0, wr0, h);

    // ---- layer 1 ----
    zero_f32<<<(NN * DIM + THREADS - 1) / THREADS, THREADS, 0, stream>>>(agg, NN * DIM);
    scatter_kernel<<<(int)((nScat + THREADS - 1) / THREADS), THREADS, 0, stream>>>(
        h, src, dst, agg, E);
    sage_gemm<false><<<(NT + 7) / 8, THREADS, 0, stream>>>(
        agg, deg, h, wl1, bl1, wr1, out);

    // ---- L2 normalize rows ----
    l2norm_kernel<<<(NN * 8 + THREADS - 1) / THREADS, THREADS, 0, stream>>>(out, NN);
}